// TreeSearch_5583457485035
// MI455X (gfx1250) — compile-verified
//
#include <hip/hip_runtime.h>
#include <stdint.h>

// ---------------------------------------------------------------------------
// Greedy MCS tree search, one workgroup (8 wave32) per batch pair.
//  * adjacency bitpacked to 512x16 u32, staged to LDS via async-to-LDS loads
//  * frontier update = 16 word ORs per graph per step (adj symmetric)
//  * cached per-row (max, argcol) with lazy invalidation -> minimal y traffic
//  * y^T staged via V_WMMA_F32_16X16X4_F32 identity trick (exact) so the
//    incremental new-column updates are coalesced
// ---------------------------------------------------------------------------

constexpr int   kB     = 128;
constexpr int   kN     = 512;
constexpr int   kW     = 16;            // 512 bits -> 16 u32 words
constexpr int   kMat   = kN * kN;       // 262144
constexpr float kThresh = 0.5f;

typedef float v2f __attribute__((ext_vector_type(2)));
typedef float v8f __attribute__((ext_vector_type(8)));

// ---------------- zero the output (harness poisons d_out) -------------------
__global__ void zero_f32(float* p, int n4) {
  int i = blockIdx.x * blockDim.x + threadIdx.x;
  float4 z = make_float4(0.f, 0.f, 0.f, 0.f);
  for (; i < n4; i += gridDim.x * blockDim.x) ((float4*)p)[i] = z;
}

// ---------------- bitpack 0/1 float adjacency into u32 masks ---------------
__global__ void bitpack(const float* __restrict__ a, uint32_t* __restrict__ bits, int n) {
  int i = blockIdx.x * blockDim.x + threadIdx.x;
  int lane = threadIdx.x & 31;
  for (; i < n; i += gridDim.x * blockDim.x) {
    uint32_t w = (uint32_t)__ballot(a[i] > 0.5f);   // wave32: 32 cols -> 1 word
    if (lane == 0) bits[i >> 5] = w;
  }
}

// ---------------- y -> y^T using WMMA f32 16x16x4, D = I4 @ B --------------
// A = identity (rows 0..3), B = 4x16 tile of y loaded in B layout; D = B but
// re-laid-out so each lane holds 4 consecutive rows of one column -> float4
// stores into y^T. Exact in f32 (x*1 + 0).
__global__ void transpose_wmma(const float* __restrict__ y, float* __restrict__ yT) {
  const int b = blockIdx.x;
  const int tid = threadIdx.x, lane = tid & 31, wave = tid >> 5;
  const float* yb = y  + (size_t)b * kMat;
  float*       tb = yT + (size_t)b * kMat;
  const int m = lane & 15;
  v2f A;
  A.x = (lane < 16) ? (m == 0 ? 1.f : 0.f) : (m == 2 ? 1.f : 0.f);  // K in VGPR0
  A.y = (lane < 16) ? (m == 1 ? 1.f : 0.f) : (m == 3 ? 1.f : 0.f);  // K in VGPR1
  // 32 col-tiles (c0 step 16) x 128 row-tiles (r0 step 4)
  for (int u = wave; u < 32 * 128; u += 8) {
    const int c0 = (u & 31) * 16;
    const int r0 = (u >> 5) * 4;
    const int rB = r0 + ((lane >> 4) << 1);   // lanes<16: K=0,1 ; lanes>=16: K=2,3
    const int cB = c0 + m;
    v2f Bf;
    Bf.x = yb[rB * kN + cB];
    Bf.y = yb[(rB + 1) * kN + cB];
    v8f C = {};
    v8f D = __builtin_amdgcn_wmma_f32_16x16x4_f32(false, A, false, Bf,
                                                  (short)0, C, false, false);
    if (lane < 16) {
      float4 st = make_float4(D[0], D[1], D[2], D[3]);
      *(float4*)(tb + (c0 + lane) * kN + r0) = st;   // yT[c][r0..r0+3]
    }
  }
}

// ---------------- wave32 argmax reduce with min-index tie-break ------------
__device__ __forceinline__ void wave_argmax(float& m, int& idx) {
#pragma unroll
  for (int off = 16; off > 0; off >>= 1) {
    float om = __shfl_xor(m,   off, 32);
    int   oi = __shfl_xor(idx, off, 32);
    if (om > m || (om == m && oi < idx)) { m = om; idx = oi; }
  }
}

// ---------------- the search -----------------------------------------------
__global__ void mcs_search(const float* __restrict__ y, const float* __restrict__ yT,
                           const uint32_t* __restrict__ gbits1,
                           const uint32_t* __restrict__ gbits2,
                           const float* __restrict__ a1g, const float* __restrict__ a2g,
                           float* __restrict__ out, int use_yt, int use_bits) {
  extern __shared__ uint32_t sm[];
  uint32_t* adjA   = sm;                               // 8192 words (32KB)
  uint32_t* adjB   = adjA + kN * kW;                   // 8192 words (32KB)
  float*    rowmax = (float*)(adjB + kN * kW);         // 512
  int*      rowarg = (int*)(rowmax + kN);              // 512
  int*      dirty  = rowarg + kN;                      // 512
  int*      newcol = dirty + kN;                       // 512
  uint32_t* f1w  = (uint32_t*)(newcol + kN);           // 16 each:
  uint32_t* f2w  = f1w + kW;
  uint32_t* of1w = f2w + kW;
  uint32_t* of2w = of1w + kW;
  uint32_t* v1w  = of2w + kW;
  uint32_t* v2w  = v1w + kW;
  uint32_t* n1w  = v2w + kW;
  uint32_t* n2w  = n1w + kW;
  float*    redv = (float*)(n2w + kW);                 // 8
  int*      redi = (int*)(redv + 8);                   // 8
  int*      cnt  = redi + 8;                           // [0]=ndirty [1]=nnew [2]=i [3]=j
  float*    gmx  = (float*)(cnt + 4);                  // 1

  const int b = blockIdx.x;
  const int tid = threadIdx.x, lane = tid & 31, wave = tid >> 5;
  const float* yb  = y  + (size_t)b * kMat;
  const float* ytb = yT + (size_t)b * kMat;            // deref only if use_yt
  float*       ob  = out + (size_t)b * kMat;

  // ---- stage adjacency bitmasks into LDS ----------------------------------
  if (use_bits) {
    const uint32_t* g1 = gbits1 + (size_t)b * kN * kW;
    const uint32_t* g2 = gbits2 + (size_t)b * kN * kW;
    uint32_t l1 = (uint32_t)(uintptr_t)adjA;           // flat low bits = LDS offset
    uint32_t l2 = (uint32_t)(uintptr_t)adjB;
    for (int o = tid * 4; o < kN * kW; o += blockDim.x * 4) {
      uint32_t la = l1 + o * 4;
      uint64_t ga = (uint64_t)(uintptr_t)(g1 + o);
      asm volatile("global_load_async_to_lds_b128 %0, %1, off"
                   :: "v"(la), "v"(ga) : "memory");
      uint32_t lb = l2 + o * 4;
      uint64_t gb = (uint64_t)(uintptr_t)(g2 + o);
      asm volatile("global_load_async_to_lds_b128 %0, %1, off"
                   :: "v"(lb), "v"(gb) : "memory");
    }
    asm volatile("s_wait_asynccnt 0" ::: "memory");
  } else {                                             // fallback: pack in-kernel
    const float* s1 = a1g + (size_t)b * kMat;
    const float* s2 = a2g + (size_t)b * kMat;
    for (int e = tid; e < kMat; e += blockDim.x) {
      uint32_t w1 = (uint32_t)__ballot(s1[e] > 0.5f);
      uint32_t w2 = (uint32_t)__ballot(s2[e] > 0.5f);
      if (lane == 0) { adjA[e >> 5] = w1; adjB[e >> 5] = w2; }
    }
  }
  for (int w = tid; w < kW; w += blockDim.x) {
    v1w[w] = 0; v2w[w] = 0; n1w[w] = 0; n2w[w] = 0;
    f1w[w] = 0xFFFFFFFFu; f2w[w] = 0xFFFFFFFFu;        // first fmask = all ones
    of1w[w] = 0u;                                      // -> all rows "new"/dirty
    of2w[w] = 0xFFFFFFFFu;                             // -> no "new" columns
  }
  __syncthreads();

  for (int iter = 0; iter < kN + 2; ++iter) {
    // ---- A: new-column list, dirty list, incremental row-max updates ------
    if (tid == 0) { cnt[0] = 0; cnt[1] = 0; }
    __syncthreads();
    for (int c = tid; c < kN; c += blockDim.x) {
      uint32_t nb = f2w[c >> 5] & ~of2w[c >> 5];
      if ((nb >> (c & 31)) & 1u) newcol[atomicAdd(&cnt[1], 1)] = c;
    }
    __syncthreads();
    const int nnew = cnt[1];
    for (int r = tid; r < kN; r += blockDim.x) {
      if (!((f1w[r >> 5] >> (r & 31)) & 1u)) continue;
      bool isdirty = !((of1w[r >> 5] >> (r & 31)) & 1u);   // newly in frontier
      if (!isdirty) {
        int a = rowarg[r];                                  // valid: row was in f1
        isdirty = !((f2w[a >> 5] >> (a & 31)) & 1u);        // cached argcol left f2
      }
      if (isdirty) {
        dirty[atomicAdd(&cnt[0], 1)] = r;
      } else {
        float m = rowmax[r]; int am = rowarg[r];
        for (int k = 0; k < nnew; ++k) {
          int c = newcol[k];
          float v = use_yt ? ytb[c * kN + r] : yb[r * kN + c];
          if (v > m || (v == m && c < am)) { m = v; am = c; }
        }
        rowmax[r] = m; rowarg[r] = am;
      }
    }
    __syncthreads();
    // ---- B: full masked rescans of dirty rows, one wave per row -----------
    const int nd = cnt[0];
    for (int d = wave; d < nd; d += 8) {
      int r = dirty[d];
      float m = -1.f; int am = 0;
#pragma unroll
      for (int k = 0; k < kW; ++k) {
        if ((f2w[k] >> lane) & 1u) {              // col = k*32+lane, coalesced
          float v = yb[r * kN + k * 32 + lane];
          if (v > m) { m = v; am = k * 32 + lane; }   // strict > keeps min col
        }
      }
      wave_argmax(m, am);
      if (lane == 0) { rowmax[r] = m; rowarg[r] = am; }
    }
    __syncthreads();
    // ---- C: global argmax over frontier rows (first-flat-index ties) ------
    float m = -2.f; int flat = 0x7FFFFFFF;
    for (int r = tid; r < kN; r += blockDim.x) {
      if (!((f1w[r >> 5] >> (r & 31)) & 1u)) continue;
      float v = rowmax[r];
      int f = r * kN + rowarg[r];
      if (v > m || (v == m && f < flat)) { m = v; flat = f; }
    }
    wave_argmax(m, flat);
    if (lane == 0) { redv[wave] = m; redi[wave] = flat; }
    __syncthreads();
    if (tid == 0) {
      float bm = -2.f; int bf = 0x7FFFFFFF;
      for (int w = 0; w < 8; ++w)
        if (redv[w] > bm || (redv[w] == bm && redi[w] < bf)) { bm = redv[w]; bf = redi[w]; }
      *gmx = bm; cnt[2] = bf >> 9; cnt[3] = bf & (kN - 1);
    }
    __syncthreads();
    if (*gmx < kThresh) break;                     // stop check BEFORE commit
    const int pi = cnt[2], pj = cnt[3];
    // ---- D: commit pick and update frontiers (adj symmetric: row==col) ----
    if (tid == 0) ob[pi * kN + pj] = 1.0f;
    if (tid < kW) { of1w[tid] = f1w[tid]; of2w[tid] = f2w[tid]; }
    __syncthreads();
    if (tid < kW) {
      uint32_t vb = v1w[tid] | ((tid == (pi >> 5)) ? (1u << (pi & 31)) : 0u);
      uint32_t nb = n1w[tid] | adjA[pi * kW + tid];
      v1w[tid] = vb; n1w[tid] = nb; f1w[tid] = nb & ~vb;
    } else if (tid < 2 * kW) {
      int w = tid - kW;
      uint32_t vb = v2w[w] | ((w == (pj >> 5)) ? (1u << (pj & 31)) : 0u);
      uint32_t nb = n2w[w] | adjB[pj * kW + w];
      v2w[w] = vb; n2w[w] = nb; f2w[w] = nb & ~vb;
    }
    __syncthreads();
  }
}

// ---------------------------------------------------------------------------
extern "C" void kernel_launch(void* const* d_in, const int* in_sizes, int n_in,
                              void* d_out, int out_size, void* d_ws, size_t ws_size,
                              hipStream_t stream) {
  const float* y  = (const float*)d_in[0];
  const float* a1 = (const float*)d_in[1];
  const float* a2 = (const float*)d_in[2];
  float* out = (float*)d_out;

  uint32_t* bits1 = (uint32_t*)d_ws;
  uint32_t* bits2 = bits1 + (size_t)kB * kN * kW;
  const size_t bits_bytes = (size_t)kB * kN * kW * 4ull * 2ull;         // 8 MB
  float* yT = (float*)((char*)d_ws + bits_bytes);
  const size_t need_yt = bits_bytes + (size_t)kB * kMat * sizeof(float); // +128 MB
  const int use_bits = ws_size >= bits_bytes ? 1 : 0;
  const int use_yt   = ws_size >= need_yt   ? 1 : 0;

  zero_f32<<<2048, 256, 0, stream>>>(out, kB * kMat / 4);
  if (use_bits) {
    bitpack<<<4096, 256, 0, stream>>>(a1, bits1, kB * kMat);
    bitpack<<<4096, 256, 0, stream>>>(a2, bits2, kB * kMat);
  }
  if (use_yt) transpose_wmma<<<kB, 256, 0, stream>>>(y, yT);

  constexpr int kSmemWords = kN * kW * 2 + kN * 4 + kW * 8 + 8 + 8 + 4 + 1;
  constexpr size_t kSmemBytes = ((size_t)kSmemWords * 4 + 255) & ~(size_t)255;
  hipFuncSetAttribute(reinterpret_cast<const void*>(mcs_search),
                      hipFuncAttributeMaxDynamicSharedMemorySize, (int)kSmemBytes);
  mcs_search<<<kB, 256, kSmemBytes, stream>>>(y, yT, bits1, bits2, a1, a2,
                                              out, use_yt, use_bits);
}